// SoftDTW_39728447488329
// MI455X (gfx1250) — compile-verified
//
#include <hip/hip_runtime.h>

typedef float v2f __attribute__((ext_vector_type(2)));
typedef float v8f __attribute__((ext_vector_type(8)));

#define BB   64
#define MM   512
#define NN   512
#define KD   64
#define BIGV 1e10f

// ---------------------------------------------------------------------------
// Kernel 1: D[b][m][n] = max(0, |x_m|^2 + |y_n|^2 - 2 * x_m . y_n)
// One wave32 computes one 16x16 output tile via V_WMMA_F32_16X16X4_F32,
// accumulating K=64 in 16 chained WMMAs. Row/col norms are computed from the
// same register operands (no extra pass over memory).
// ---------------------------------------------------------------------------
__global__ __launch_bounds__(256) void pairdist_wmma(
    const float* __restrict__ x, const float* __restrict__ y,
    float* __restrict__ D)
{
    const int lane = threadIdx.x & 31;
    const int wave = blockIdx.x * (blockDim.x >> 5) + (threadIdx.x >> 5);
    const int b  = wave >> 10;          // 1024 tiles per batch
    const int t  = wave & 1023;
    const int tm = t >> 5;              // tile row 0..31
    const int tn = t & 31;              // tile col 0..31

    // A: 16x4 f32 tile layout -> lane holds row (lane&15), k = 2*(lane>>4)+{0,1}
    const float* xrow = x + (size_t)b * MM * KD + (size_t)(tm * 16 + (lane & 15)) * KD;
    const float* ycol = y + (size_t)b * NN * KD + (size_t)(tn * 16 + (lane & 15)) * KD;
    const int kb = (lane >> 4) * 2;

    v8f acc = {};
    float xs = 0.0f, ys = 0.0f;
#pragma unroll
    for (int k0 = 0; k0 < KD; k0 += 4) {
        v2f a  = *(const v2f*)(xrow + k0 + kb);
        v2f bb = *(const v2f*)(ycol + k0 + kb);
        xs += a.x * a.x + a.y * a.y;
        ys += bb.x * bb.x + bb.y * bb.y;
        // D = A * B + C   (A = x tile 16x4, B = y^T tile 4x16)
        acc = __builtin_amdgcn_wmma_f32_16x16x4_f32(
                  false, a, false, bb, (short)0, acc, false, false);
    }
    // complete norms: lane L and L^16 cover complementary k-offsets of same row/col
    xs += __shfl_xor(xs, 16, 32);
    ys += __shfl_xor(ys, 16, 32);

    // C/D layout: lane L, vgpr r -> element [M = r + 8*(L>=16)][N = L&15]
    float* Dp = D + (size_t)b * MM * NN + (size_t)(tm * 16) * NN
                  + (size_t)(tn * 16) + (lane & 15);
    const int mhi = (lane >> 4) * 8;
#pragma unroll
    for (int r = 0; r < 8; ++r) {
        int ml   = r + mhi;                       // local output row 0..15
        float xn = __shfl(xs, ml, 32);            // norm of that row (held by lane ml)
        float v  = xn + ys - 2.0f * acc[r];
        Dp[(size_t)ml * NN] = fmaxf(v, 0.0f);
    }
}

// ---------------------------------------------------------------------------
// Kernel 2: soft-DTW anti-diagonal DP. One block per batch, thread i-1 owns
// row i. Three rolling (M+1)-rows in LDS; read set {v1,v2} and write set {vd}
// are disjoint each step, so ONE barrier per diagonal suffices.
// D loads are per-thread sequential in j (next-diagonal load is pipelined).
// ---------------------------------------------------------------------------
__global__ __launch_bounds__(512) void softdtw_dp(
    const float* __restrict__ D, float* __restrict__ partial)
{
    __shared__ float buf[3][MM + 1];
    const int b   = blockIdx.x;
    const int tid = threadIdx.x;
    const int i   = tid + 1;                      // DP row index 1..M

    for (int idx = tid; idx <= MM; idx += blockDim.x) {
        buf[0][idx] = (idx == 0) ? 0.0f : BIGV;   // v0
        buf[1][idx] = BIGV;                       // v1
    }
    __syncthreads();

    const float* Drow = D + (size_t)b * MM * NN + (size_t)tid * NN;

    float* v2 = buf[0];
    float* v1 = buf[1];
    float* vd = buf[2];

    const float cexp = 14.4269504088896340736f;   // (1/gamma)*log2(e), gamma=0.1
    const float clog = 0.069314718055994530942f;  // gamma*ln(2)

    // software-pipelined D load for d=2
    int jn = 2 - i;
    float dval = (jn >= 1 && jn <= NN) ? Drow[jn - 1] : 0.0f;

    for (int d = 2; d <= MM + NN; ++d) {
        const float dcur = dval;
        const int j = d - i;
        jn = j + 1;                               // preload next diagonal's element
        dval = (jn >= 1 && jn <= NN) ? Drow[jn - 1] : 0.0f;

        float r0 = v2[i - 1];
        float r1 = v1[i - 1];
        float r2 = v1[i];
        float mn = fminf(r0, fminf(r1, r2));
        float s  = __builtin_exp2f((mn - r0) * cexp)
                 + __builtin_exp2f((mn - r1) * cexp)
                 + __builtin_exp2f((mn - r2) * cexp);
        float softmin = mn - clog * __builtin_log2f(s);
        float val = (j >= 1 && j <= NN) ? (dcur + softmin) : BIGV;

        vd[i] = val;
        if (tid == 0) vd[0] = BIGV;
        __syncthreads();                          // single barrier per diagonal

        float* tmp = v2; v2 = v1; v1 = vd; vd = tmp;
    }
    if (tid == 0) partial[b] = v1[MM];            // last-written row, element M
}

// ---------------------------------------------------------------------------
// Kernel 3: mean over 64 per-batch results.
// ---------------------------------------------------------------------------
__global__ void reduce_mean(const float* __restrict__ partial, float* __restrict__ out)
{
    __shared__ float s[BB];
    const int t = threadIdx.x;
    s[t] = partial[t];
    __syncthreads();
    if (t == 0) {
        float acc = 0.0f;
        for (int k = 0; k < BB; ++k) acc += s[k];
        out[0] = acc * (1.0f / (float)BB);
    }
}

extern "C" void kernel_launch(void* const* d_in, const int* in_sizes, int n_in,
                              void* d_out, int out_size, void* d_ws, size_t ws_size,
                              hipStream_t stream)
{
    const float* x = (const float*)d_in[0];
    const float* y = (const float*)d_in[1];

    float* D       = (float*)d_ws;                     // 64*512*512 floats = 64 MB
    float* partial = D + (size_t)BB * MM * NN;         // +64 floats

    // Phase 1: 65536 tile-waves, 8 waves per 256-thread block
    pairdist_wmma<<<8192, 256, 0, stream>>>(x, y, D);
    // Phase 2: one block per batch, 512 threads (16 wave32)
    softdtw_dp<<<BB, MM, 0, stream>>>(D, partial);
    // Phase 3: scalar mean
    reduce_mean<<<1, BB, 0, stream>>>(partial, (float*)d_out);
}